// ConvolutionLayer_56599079027373
// MI455X (gfx1250) — compile-verified
//
#include <hip/hip_runtime.h>
#include <hip/hip_bf16.h>

// Implicit-GEMM 5x5 VALID conv for MI455X (gfx1250, wave32) on the dedicated
// FP32 matrix path V_WMMA_F32_16X16X4_F32.
//
// GEMM view:  out[64 x (16*220*220)] = W[64 x 800] * im2col(x)[800 x N]
// Workgroup (8 waves) = ALL 64 filters x 256 columns at one (batch, oy).
// All weights (64 x 28 padded taps x 32 ch = 229 KB) are staged ONCE into the
// 320 KB WGP LDS (dynamic LDS; CDNA5-only capacity); the channel loop only
// double-buffers the 5x260 x patch.
//
// Wave tile = 64(M) x 32(N): 4 A-frags x 2 B-frags -> 8 WMMAs per 6 DS loads
// (0.75 ds/wmma), 8 independent accumulator chains interleaved on the pipe.
//
// K-order permutation (closed form -> constant DS offsets, no tables):
//   K = 4s+j, s=0..4 : tap (row s, kx = 2*half + j%2)  -> B pair = (+0,+1)
//   K = 20..23       : taps (row 2*half + j, kx=4)     -> B pair = (+0,+XCOLS)
//   K = 24..27       : tap (4,4) then zero-pads        -> A weight 0 for pads

typedef float v2f __attribute__((ext_vector_type(2)));
typedef float v8f __attribute__((ext_vector_type(8)));

#define BATCH   16
#define C_IN    32
#define HIN     224
#define N_FILT  64
#define KSZ     5
#define HOUT    220            // 224 - 5 + 1
#define TAPS    25             // per-channel taps (5x5)
#define TAPS_P  28             // padded to multiple of WMMA K=4 (pad weights=0)
#define TILE_N  256            // output columns per workgroup (all of 220, padded)
#define XCOLS   260            // TILE_N + KSZ - 1 patch width
#define W_ELEMS (N_FILT * TAPS_P * C_IN)    // 57344 floats = 229 KB
#define X_ELEMS (KSZ * XCOLS)               // 1300 floats per buffer
#define SMEM_BYTES ((W_ELEMS + 2 * X_ELEMS) * 4)   // 239.8 KB < 320 KB WGP LDS

#define WMMAF32(A, B, C) \
    __builtin_amdgcn_wmma_f32_16x16x4_f32(false, (A), false, (B), (short)0, (C), false, false)

// 8 WMMAs on a 64x32 tile: interleave accumulators so consecutive WMMAs never
// share a D register (keeps the f32 matrix pipe free of D->C stalls).
#define DO8(a0, a1, a2, a3, b0, b1)            \
    acc00 = WMMAF32(a0, b0, acc00);            \
    acc10 = WMMAF32(a1, b0, acc10);            \
    acc20 = WMMAF32(a2, b0, acc20);            \
    acc30 = WMMAF32(a3, b0, acc30);            \
    acc01 = WMMAF32(a0, b1, acc01);            \
    acc11 = WMMAF32(a1, b1, acc11);            \
    acc21 = WMMAF32(a2, b1, acc21);            \
    acc31 = WMMAF32(a3, b1, acc31);

__global__ __launch_bounds__(256)
void conv5x5_wmma_f32(const float* __restrict__ x,
                      const float* __restrict__ w,
                      const float* __restrict__ bias,
                      float* __restrict__ out) {
    extern __shared__ float smem[];   // [0,W_ELEMS): weights | 2 x-patch buffers

    const int tid  = threadIdx.x;
    const int lane = tid & 31;
    const int wave = tid >> 5;       // 0..7 : 32-column tile index
    const int half = lane >> 4;      // 0: lanes 0-15, 1: lanes 16-31
    const int l16  = lane & 15;

    const int oy = blockIdx.x;             // output row
    const int b  = blockIdx.y;             // batch

    // ---- One-time weight stage: [c][64 filters][28 permuted taps]
    for (int i = tid; i < W_ELEMS; i += 256) {
        const int c   = i / (N_FILT * TAPS_P);
        const int rem = i - c * (N_FILT * TAPS_P);
        const int f   = rem / TAPS_P;
        const int t   = rem - f * TAPS_P;
        float v = 0.0f;
        if (t < 20)                 // within-row pairs: tap (t/4, t%4)
            v = w[(f * C_IN + c) * TAPS + (t >> 2) * KSZ + (t & 3)];
        else if (t < TAPS)          // kx=4 column: tap (t-20, 4)
            v = w[(f * C_IN + c) * TAPS + (t - 20) * KSZ + 4];
        smem[i] = v;                // t = 25..27 stay 0 (zero-pad)
    }

    // ---- x patch stager: 5 rows x 260 cols, edge-clamped (offset-indexed buffer)
    auto load_x = [&](int c, int buf) {
        const int base = W_ELEMS + buf * X_ELEMS;
        for (int i = tid; i < X_ELEMS; i += 256) {
            const int dy  = i / XCOLS;
            const int dx  = i - dy * XCOLS;
            const int col = (dx > HIN - 1) ? (HIN - 1) : dx;  // feeds masked cols only
            smem[base + i] = x[((b * C_IN + c) * HIN + (oy + dy)) * HIN + col];
        }
    };

    load_x(0, 0);
    __syncthreads();

    // 64x32 wave tile: 8 independent 16x16 f32 C/D accumulators
    // (VGPR r holds M = m*16 + half*8 + r, N = l16 within subtile).
    v8f acc00 = {}, acc01 = {}, acc10 = {}, acc11 = {};
    v8f acc20 = {}, acc21 = {}, acc30 = {}, acc31 = {};

    const int ncol = wave * 32 + l16;      // this lane's column base (0..255)

    for (int c = 0; c < C_IN; ++c) {
        const int cur = c & 1;
        if (c + 1 < C_IN) load_x(c + 1, cur ^ 1);   // stage next channel

        // global_prefetch_b8: pull channel c+2's x rows one iteration ahead
        if (c + 2 < C_IN && tid < 40) {
            const int dy  = tid >> 3;            // 5 rows
            const int seg = tid & 7;             // 128B cacheline segments
            if (seg * 128 < HIN * 4)             // 7 lines cover a 896B row
                __builtin_prefetch(
                    (const char*)&x[((b * C_IN + (c + 2)) * HIN + (oy + dy)) * HIN]
                        + seg * 128, 0, 3);
        }

        const int wbase = c * (N_FILT * TAPS_P) + l16 * TAPS_P;   // A, m=0 subtile
        const int xbase = W_ELEMS + cur * X_ELEMS + ncol;         // B, lane column

        // ---- steps 0..4: K=4s..4s+3 = row s, kx = 2*half + {0,1}
#pragma unroll
        for (int s = 0; s < 5; ++s) {
            const int t0 = s * 4 + half * 2;
            v2f a0, a1, a2, a3, b0, b1;
            a0.x = smem[wbase + t0];                   a0.y = smem[wbase + t0 + 1];
            a1.x = smem[wbase + 16 * TAPS_P + t0];     a1.y = smem[wbase + 16 * TAPS_P + t0 + 1];
            a2.x = smem[wbase + 32 * TAPS_P + t0];     a2.y = smem[wbase + 32 * TAPS_P + t0 + 1];
            a3.x = smem[wbase + 48 * TAPS_P + t0];     a3.y = smem[wbase + 48 * TAPS_P + t0 + 1];
            const int bo = xbase + s * XCOLS + half * 2;
            b0.x = smem[bo];       b0.y = smem[bo + 1];        // 2addr offset1:1
            b1.x = smem[bo + 16];  b1.y = smem[bo + 17];
            DO8(a0, a1, a2, a3, b0, b1)
        }

        // ---- step 5: K=20..23 = taps (2*half + {0,1}, kx=4)
        {
            const int t0 = 20 + half * 2;
            v2f a0, a1, a2, a3, b0, b1;
            a0.x = smem[wbase + t0];                   a0.y = smem[wbase + t0 + 1];
            a1.x = smem[wbase + 16 * TAPS_P + t0];     a1.y = smem[wbase + 16 * TAPS_P + t0 + 1];
            a2.x = smem[wbase + 32 * TAPS_P + t0];     a2.y = smem[wbase + 32 * TAPS_P + t0 + 1];
            a3.x = smem[wbase + 48 * TAPS_P + t0];     a3.y = smem[wbase + 48 * TAPS_P + t0 + 1];
            const int bo = xbase + (half * 2) * XCOLS + 4;
            b0.x = smem[bo];       b0.y = smem[bo + XCOLS];    // adjacent rows
            b1.x = smem[bo + 16];  b1.y = smem[bo + XCOLS + 16];
            DO8(a0, a1, a2, a3, b0, b1)
        }

        // ---- step 6: K=24 = tap (4,4); K=25..27 zero-pad (A weight 0, B aliased)
        {
            const int t0 = 24 + half * 2;
            v2f a0, a1, a2, a3, b0, b1;
            a0.x = smem[wbase + t0];                   a0.y = smem[wbase + t0 + 1];
            a1.x = smem[wbase + 16 * TAPS_P + t0];     a1.y = smem[wbase + 16 * TAPS_P + t0 + 1];
            a2.x = smem[wbase + 32 * TAPS_P + t0];     a2.y = smem[wbase + 32 * TAPS_P + t0 + 1];
            a3.x = smem[wbase + 48 * TAPS_P + t0];     a3.y = smem[wbase + 48 * TAPS_P + t0 + 1];
            const int bo = xbase + 4 * XCOLS + 4;
            v2f t; t.x = smem[bo]; t.y = smem[bo + 16];
            b0.x = t.x;  b0.y = t.x;   // pad K: value irrelevant (A weight = 0)
            b1.x = t.y;  b1.y = t.y;
            DO8(a0, a1, a2, a3, b0, b1)
        }

        __syncthreads();  // cur x-buffer fully consumed; next buffer fully staged
    }

    // Scatter the 64x32 wave tile. Predication only here -> EXEC was all-1s
    // for every WMMA.
#pragma unroll
    for (int r = 0; r < 8; ++r) {
        const int fr  = half * 8 + r;
        const int ox0 = ncol;
        const int ox1 = ncol + 16;
        if (ox0 < HOUT) {
            out[((b * N_FILT + fr     ) * HOUT + oy) * HOUT + ox0] = acc00[r] + bias[fr     ];
            out[((b * N_FILT + fr + 16) * HOUT + oy) * HOUT + ox0] = acc10[r] + bias[fr + 16];
            out[((b * N_FILT + fr + 32) * HOUT + oy) * HOUT + ox0] = acc20[r] + bias[fr + 32];
            out[((b * N_FILT + fr + 48) * HOUT + oy) * HOUT + ox0] = acc30[r] + bias[fr + 48];
        }
        if (ox1 < HOUT) {
            out[((b * N_FILT + fr     ) * HOUT + oy) * HOUT + ox1] = acc01[r] + bias[fr     ];
            out[((b * N_FILT + fr + 16) * HOUT + oy) * HOUT + ox1] = acc11[r] + bias[fr + 16];
            out[((b * N_FILT + fr + 32) * HOUT + oy) * HOUT + ox1] = acc21[r] + bias[fr + 32];
            out[((b * N_FILT + fr + 48) * HOUT + oy) * HOUT + ox1] = acc31[r] + bias[fr + 48];
        }
    }
}

extern "C" void kernel_launch(void* const* d_in, const int* in_sizes, int n_in,
                              void* d_out, int out_size, void* d_ws, size_t ws_size,
                              hipStream_t stream) {
    const float* x    = (const float*)d_in[0];
    const float* wgt  = (const float*)d_in[1];
    const float* bias = (const float*)d_in[2];
    float* out        = (float*)d_out;

    dim3 grid(HOUT, BATCH);   // (220, 16)
    dim3 block(256);
    hipLaunchKernelGGL(conv5x5_wmma_f32, grid, block, SMEM_BYTES, stream,
                       x, wgt, bias, out);
}